// texture_feature_extractor_18193481466432
// MI455X (gfx1250) — compile-verified
//
#include <hip/hip_runtime.h>
#include <hip/hip_bf16.h>
#include <float.h>

typedef __attribute__((ext_vector_type(2))) float v2f;
typedef __attribute__((ext_vector_type(8))) float v8f;

#define OUT_STRIDE 791
#define OFS_PIX          0
#define OFS_MEAN_MAG     6
#define OFS_MEAN_REAL   22
#define OFS_MARG        38
#define OFS_HP_VAR      48
#define OFS_MAG_AC      49
#define OFS_LOW_AC     449
#define OFS_HP_AC      574
#define OFS_MAG_WITHIN 599
#define OFS_REAL_WITHIN 623
#define OFS_MAG_ACROSS 647
#define OFS_RI_ACROSS  695

struct Ptrs {
  const float* images; const float* hp;
  const float* l[5];
  const float* cr[4]; const float* ci[4];
  const float* ur[3]; const float* ui[3];
};

__device__ __forceinline__ float gaussw(int yy, int xx, float inv_n) {
  float cy = ((float)yy + 0.5f) * inv_n - 0.5f;
  float cx = ((float)xx + 0.5f) * inv_n - 0.5f;
  return __expf(-(cx * cx + cy * cy) * 50.0f); // 1/(2*sigma^2), sigma=0.1
}

__device__ __forceinline__ float blockReduceSum(float v, float* red) {
  int tid = threadIdx.x;
  red[tid] = v; __syncthreads();
  for (int st = 128; st > 0; st >>= 1) { if (tid < st) red[tid] += red[tid + st]; __syncthreads(); }
  float r = red[0]; __syncthreads();
  return r;
}
__device__ __forceinline__ float blockReduceMin(float v, float* red) {
  int tid = threadIdx.x;
  red[tid] = v; __syncthreads();
  for (int st = 128; st > 0; st >>= 1) { if (tid < st) red[tid] = fminf(red[tid], red[tid + st]); __syncthreads(); }
  float r = red[0]; __syncthreads();
  return r;
}
__device__ __forceinline__ float blockReduceMax(float v, float* red) {
  int tid = threadIdx.x;
  red[tid] = v; __syncthreads();
  for (int st = 128; st > 0; st >>= 1) { if (tid < st) red[tid] = fmaxf(red[tid], red[tid + st]); __syncthreads(); }
  float r = red[0]; __syncthreads();
  return r;
}

// ---------------------------------------------------------------------------
// Pass 1: weighted means / moments / min-max. 75 tasks per batch element.
//   0: images -> pix (m,v,s,k,min,max)     1: hp -> hp_var
//   2..6: l0..l4 -> marg (s,k)
//   7..22: mag(ff,o)  -> mean_mag + ws slot0     23..38: real -> mean_real + slot1
//   39..50: umag -> slot2   51..62: ur -> slot3   63..74: ui -> slot4
// ---------------------------------------------------------------------------
__global__ void stats_kernel(Ptrs P, float* __restrict__ out, float* __restrict__ wsMeans) {
  __shared__ float red[256];
  const int task = blockIdx.x;
  const int b = blockIdx.y;
  const int CSZ[4] = {30, 60, 120, 240};
  const int LSZ[5] = {15, 30, 60, 120, 240};

  const float* p0 = nullptr; const float* p1 = nullptr;
  int n = 240, mode = 0, ff = 0, o = 0, lev = 0;

  if (task == 0)      { p0 = P.images + (size_t)b * 57600; }
  else if (task == 1) { p0 = P.hp + (size_t)b * 57600; }
  else if (task < 7)  { lev = task - 2; n = LSZ[lev]; p0 = P.l[lev] + (size_t)b * n * n; }
  else if (task < 23) { int i = task - 7;  ff = i >> 2; o = i & 3; n = CSZ[ff];
                        size_t ofs = ((size_t)b * 4 + o) * (size_t)n * n;
                        p0 = P.cr[ff] + ofs; p1 = P.ci[ff] + ofs; mode = 1; }
  else if (task < 39) { int i = task - 23; ff = i >> 2; o = i & 3; n = CSZ[ff];
                        size_t ofs = ((size_t)b * 4 + o) * (size_t)n * n;
                        p0 = P.cr[ff] + ofs; }
  else if (task < 51) { int i = task - 39; ff = (i >> 2) + 1; o = i & 3; n = CSZ[ff];
                        size_t ofs = ((size_t)b * 4 + o) * (size_t)n * n;
                        p0 = P.ur[ff - 1] + ofs; p1 = P.ui[ff - 1] + ofs; mode = 1; }
  else if (task < 63) { int i = task - 51; ff = (i >> 2) + 1; o = i & 3; n = CSZ[ff];
                        size_t ofs = ((size_t)b * 4 + o) * (size_t)n * n;
                        p0 = P.ur[ff - 1] + ofs; }
  else                { int i = task - 63; ff = (i >> 2) + 1; o = i & 3; n = CSZ[ff];
                        size_t ofs = ((size_t)b * 4 + o) * (size_t)n * n;
                        p0 = P.ui[ff - 1] + ofs; }

  const int N = n * n;
  const float inv_n = 1.0f / (float)n;

  float sw = 0.f, swx = 0.f, mn = FLT_MAX, mx = -FLT_MAX;
  for (int i = threadIdx.x; i < N; i += 256) {
    int yy = i / n, xx = i - yy * n;
    float w = gaussw(yy, xx, inv_n);
    float x;
    if (mode) { float a = p0[i], c = p1[i]; x = sqrtf(a * a + c * c); }
    else      { x = p0[i]; }
    sw += w; swx += w * x;
    mn = fminf(mn, x); mx = fmaxf(mx, x);
  }
  sw = blockReduceSum(sw, red);
  swx = blockReduceSum(swx, red);
  float m = swx / sw;

  float* ob = out + (size_t)b * OUT_STRIDE;

  if (task >= 7) {
    if (threadIdx.x == 0) {
      size_t wsi = (((size_t)b * 4 + ff) * 4 + o) * 5;
      if (task < 23)      { ob[OFS_MEAN_MAG  + ff * 4 + o] = m; wsMeans[wsi + 0] = m; }
      else if (task < 39) { ob[OFS_MEAN_REAL + ff * 4 + o] = m; wsMeans[wsi + 1] = m; }
      else if (task < 51) { wsMeans[wsi + 2] = m; }
      else if (task < 63) { wsMeans[wsi + 3] = m; }
      else                { wsMeans[wsi + 4] = m; }
    }
    return;
  }

  // higher moments (tasks 0..6, all direct maps)
  float s2 = 0.f, s3 = 0.f, s4 = 0.f;
  for (int i = threadIdx.x; i < N; i += 256) {
    int yy = i / n, xx = i - yy * n;
    float w = gaussw(yy, xx, inv_n);
    float d = p0[i] - m, d2 = d * d;
    s2 += w * d2; s3 += w * d2 * d; s4 += w * d2 * d2;
  }
  s2 = blockReduceSum(s2, red);
  s3 = blockReduceSum(s3, red);
  s4 = blockReduceSum(s4, red);
  if (task == 0) { mn = blockReduceMin(mn, red); mx = blockReduceMax(mx, red); }
  if (threadIdx.x == 0) {
    float v = s2 / sw;
    float vs = fmaxf(v, 1e-12f);
    float sk = (s3 / sw) / (vs * sqrtf(vs));
    float ku = (s4 / sw) / (vs * vs);
    if (task == 0)      { ob[0] = m; ob[1] = v; ob[2] = sk; ob[3] = ku; ob[4] = mn; ob[5] = mx; }
    else if (task == 1) { ob[OFS_HP_VAR] = v; }
    else                { ob[OFS_MARG + lev * 2] = sk; ob[OFS_MARG + lev * 2 + 1] = ku; }
  }
}

// ---------------------------------------------------------------------------
// Pass 2: windowed circular autocorrelation over center bbox, direct lags.
//   22 tasks: 0..4 low levels, 5 hp, 6..21 mag(ff,o)
// ---------------------------------------------------------------------------
__global__ void autocorr_kernel(Ptrs P, float* __restrict__ out) {
  __shared__ float red[256];
  __shared__ float ptile[96 * 96];
  const int task = blockIdx.x;
  const int b = blockIdx.y;
  const int LSZ[5] = {15, 30, 60, 120, 240};
  const int LOP[5] = {3, 3, 5, 7, 7};
  const int CSZ[4] = {30, 60, 120, 240};
  const int COP[4] = {3, 5, 7, 7};

  const float* p0 = nullptr; const float* p1 = nullptr; int mode = 0;
  int n, outp, obase;
  if (task < 5)       { n = LSZ[task]; outp = LOP[task];
                        p0 = P.l[task] + (size_t)b * n * n; obase = OFS_LOW_AC + task * 25; }
  else if (task == 5) { n = 240; outp = 7; p0 = P.hp + (size_t)b * 57600; obase = OFS_HP_AC; }
  else                { int i = task - 6; int ff = i >> 2; int o = i & 3;
                        n = CSZ[ff]; outp = COP[ff];
                        size_t ofs = ((size_t)b * 4 + o) * (size_t)n * n;
                        p0 = P.cr[ff] + ofs; p1 = P.ci[ff] + ofs; mode = 1;
                        obase = OFS_MAG_AC + (ff * 4 + o) * 25; }

  const int half = n / 5;                 // ceil(0.2*n) for these sizes
  const int c = (n == 15) ? 8 : (n / 2);  // np.round(0.5*n)
  const int y0 = c - half;
  const int s = 2 * half;
  const int S2 = s * s;
  const float inv_n = 1.0f / (float)n;

  float sw = 0.f, swx = 0.f;
  for (int i = threadIdx.x; i < S2; i += 256) {
    int yy = y0 + i / s, xx = y0 + i % s;
    float w = gaussw(yy, xx, inv_n);
    float x;
    if (mode) { float a = p0[yy * n + xx], cc = p1[yy * n + xx]; x = sqrtf(a * a + cc * cc); }
    else      { x = p0[yy * n + xx]; }
    ptile[i] = x;
    sw += w; swx += w * x;
  }
  __syncthreads();
  sw = blockReduceSum(sw, red);
  swx = blockReduceSum(swx, red);
  float pm = swx / sw;
  for (int i = threadIdx.x; i < S2; i += 256) {
    int yy = y0 + i / s, xx = y0 + i % s;
    ptile[i] = (ptile[i] - pm) * gaussw(yy, xx, inv_n);
  }
  __syncthreads();

  const int h = outp >> 1;
  const int nuni = (outp * outp + 1) >> 1;
  float* ob = out + (size_t)b * OUT_STRIDE + obase;
  for (int t = 0; t < 25; t++) {
    if (t < nuni) {
      int dy = t / outp - h;
      int dx = t % outp - h;
      float acc = 0.f;
      for (int i = threadIdx.x; i < S2; i += 256) {
        int yy = i / s, xx = i - yy * s;
        int y2 = yy + dy; if (y2 < 0) y2 += s; else if (y2 >= s) y2 -= s;
        int x2 = xx + dx; if (x2 < 0) x2 += s; else if (x2 >= s) x2 -= s;
        acc += ptile[i] * ptile[y2 * s + x2];
      }
      acc = blockReduceSum(acc, red);
      if (threadIdx.x == 0) ob[t] = acc;
    } else {
      if (threadIdx.x == 0) ob[t] = 0.f;  // jnp.pad zeros
    }
  }
}

// ---------------------------------------------------------------------------
// Pass 3: all cross-correlation matrices via V_WMMA_F32_16X16X4_F32,
// templated on HAS_U so the ff==0 variant has a branch-free 8-WMMA inner loop
// and the ff>0 variant a branch-free 16-WMMA inner loop.
//   A rows (x w): [magc 0..3 | realc 4..7 | zero 8..15]
//   B1 cols:      [magc 0..3 | realc 4..7 | umagc 8..11 | urc 12..15]
//   B2 cols:      [uic 0..3 | zero 4..15]
//   -> cm=C1[0:4,0:4], cr=C1[4:8,4:8], mag_across=C1[0:4,8:12],
//      rr=C1[4:8,12:16], ri=C2[4:8,0:4]
// ---------------------------------------------------------------------------
template <bool HAS_U>
__global__ void wcross_wmma_kernel(Ptrs P, const float* __restrict__ wsMeans,
                                   float* __restrict__ out) {
  __shared__ float Atile[8][16][33];
  __shared__ float B1tile[8][16][33];
  __shared__ float B2tile[8][16][33];
  __shared__ float Cacc[2][256];

  const int ff = HAS_U ? (int)blockIdx.x + 1 : 0;
  const int b = blockIdx.y;
  const int CSZ[4] = {30, 60, 120, 240};
  const int n = CSZ[ff];
  const int K = n * n;
  const float inv_n = 1.0f / (float)n;

  const float* cr = P.cr[ff] + (size_t)b * 4 * K;
  const float* ci = P.ci[ff] + (size_t)b * 4 * K;
  const float* ur = HAS_U ? (P.ur[ff - 1] + (size_t)b * 4 * K) : nullptr;
  const float* ui = HAS_U ? (P.ui[ff - 1] + (size_t)b * 4 * K) : nullptr;

  // zero all LDS state (rows 8..15 of A, 8..15 of B1 when !HAS_U, 4..15 of B2)
  for (int i = threadIdx.x; i < 8 * 16 * 33; i += 256) {
    ((float*)Atile)[i] = 0.f; ((float*)B1tile)[i] = 0.f;
    if (HAS_U) ((float*)B2tile)[i] = 0.f;
  }
  for (int i = threadIdx.x; i < 512; i += 256) ((float*)Cacc)[i] = 0.f;
  __syncthreads();

  float wmMag[4], wmReal[4], wmUmag[4], wmUr[4], wmUi[4];
#pragma unroll
  for (int o = 0; o < 4; o++) {
    const float* base = wsMeans + (((size_t)b * 4 + ff) * 4 + o) * 5;
    wmMag[o] = base[0]; wmReal[o] = base[1];
    if (HAS_U) { wmUmag[o] = base[2]; wmUr[o] = base[3]; wmUi[o] = base[4]; }
  }

  const int wave = threadIdx.x >> 5;
  const int lane = threadIdx.x & 31;
  const int m = lane & 15;        // A row / B col for this lane
  const int hi = lane >> 4;       // K-half selector

  float (*A)[33]  = Atile[wave];
  float (*B1)[33] = B1tile[wave];
  float (*B2)[33] = B2tile[wave];

  v8f C1 = {}; v8f C2 = {};
  const int chunks = (K + 31) / 32;

  for (int cch = wave; cch < chunks; cch += 8) {
    const int kp = cch * 32 + lane;
    const bool inb = (kp < K);
    float wpix = 0.f;
    if (inb) {
      int yy = kp / n, xx = kp - yy * n;
      wpix = gaussw(yy, xx, inv_n);
    }
#pragma unroll
    for (int o = 0; o < 4; o++) {
      float crv = inb ? cr[(size_t)o * K + kp] : 0.f;
      float civ = inb ? ci[(size_t)o * K + kp] : 0.f;
      float mg = sqrtf(crv * crv + civ * civ);
      float magc  = inb ? (mg - wmMag[o])   : 0.f;
      float realc = inb ? (crv - wmReal[o]) : 0.f;
      A[o][lane]      = magc * wpix;
      A[4 + o][lane]  = realc * wpix;
      B1[o][lane]     = magc;
      B1[4 + o][lane] = realc;
      if (HAS_U) {
        float urv = inb ? ur[(size_t)o * K + kp] : 0.f;
        float uiv = inb ? ui[(size_t)o * K + kp] : 0.f;
        float um = sqrtf(urv * urv + uiv * uiv);
        B1[8 + o][lane]  = inb ? (um - wmUmag[o]) : 0.f;
        B1[12 + o][lane] = inb ? (urv - wmUr[o])  : 0.f;
        B2[o][lane]      = inb ? (uiv - wmUi[o])  : 0.f;
      }
    }
    asm volatile("s_wait_dscnt 0" ::: "memory");  // wave-private tile write -> read
#pragma unroll
    for (int kk = 0; kk < 32; kk += 4) {
      const int k0 = kk + 2 * hi;   // VGPR j of A/B holds K = j + 2*hi
      v2f a;  a.x  = A[m][k0];  a.y  = A[m][k0 + 1];
      v2f b1; b1.x = B1[m][k0]; b1.y = B1[m][k0 + 1];
      C1 = __builtin_amdgcn_wmma_f32_16x16x4_f32(false, a, false, b1,
                                                 (short)0, C1, false, false);
      if (HAS_U) {
        v2f b2; b2.x = B2[m][k0]; b2.y = B2[m][k0 + 1];
        C2 = __builtin_amdgcn_wmma_f32_16x16x4_f32(false, a, false, b2,
                                                   (short)0, C2, false, false);
      }
    }
    asm volatile("" ::: "memory");
  }

  // merge the 8 waves' accumulators: C VGPR r, lane L -> M=r+8*hi, N=L&15
#pragma unroll
  for (int r = 0; r < 8; r++) {
    atomicAdd(&Cacc[0][(r + 8 * hi) * 16 + m], C1[r]);
    if (HAS_U) atomicAdd(&Cacc[1][(r + 8 * hi) * 16 + m], C2[r]);
  }
  __syncthreads();

  if (threadIdx.x == 0) {
    const float scale = 1.0f / (float)K;
    float* ob = out + (size_t)b * OUT_STRIDE;
    const int ti[6] = {0, 0, 0, 1, 1, 2};
    const int tj[6] = {1, 2, 3, 2, 3, 3};
#pragma unroll
    for (int t = 0; t < 6; t++) {
      ob[OFS_MAG_WITHIN  + ff * 6 + t] = Cacc[0][ti[t] * 16 + tj[t]] * scale;
      ob[OFS_REAL_WITHIN + ff * 6 + t] = Cacc[0][(4 + ti[t]) * 16 + 4 + tj[t]] * scale;
    }
    if (HAS_U) {
      const int g = ff - 1;
      for (int o = 0; o < 4; o++)
        for (int pq = 0; pq < 4; pq++) {
          ob[OFS_MAG_ACROSS + g * 16 + o * 4 + pq] = Cacc[0][o * 16 + 8 + pq] * scale;
          ob[OFS_RI_ACROSS + g * 32 + o * 4 + pq]      = Cacc[0][(4 + o) * 16 + 12 + pq] * scale;
          ob[OFS_RI_ACROSS + g * 32 + 16 + o * 4 + pq] = Cacc[1][(4 + o) * 16 + pq] * scale;
        }
    }
  }
}

extern "C" void kernel_launch(void* const* d_in, const int* in_sizes, int n_in,
                              void* d_out, int out_size, void* d_ws, size_t ws_size,
                              hipStream_t stream) {
  Ptrs P;
  P.images = (const float*)d_in[0];
  P.hp     = (const float*)d_in[1];
  for (int i = 0; i < 5; i++) P.l[i]  = (const float*)d_in[2 + i];
  for (int i = 0; i < 4; i++) P.cr[i] = (const float*)d_in[7 + i];
  for (int i = 0; i < 4; i++) P.ci[i] = (const float*)d_in[11 + i];
  for (int i = 0; i < 3; i++) P.ur[i] = (const float*)d_in[15 + i];
  for (int i = 0; i < 3; i++) P.ui[i] = (const float*)d_in[18 + i];

  const int B = in_sizes[0] / (240 * 240);
  float* out = (float*)d_out;
  float* ws  = (float*)d_ws;

  stats_kernel<<<dim3(75, B), 256, 0, stream>>>(P, out, ws);
  autocorr_kernel<<<dim3(22, B), 256, 0, stream>>>(P, out);
  wcross_wmma_kernel<false><<<dim3(1, B), 256, 0, stream>>>(P, ws, out);
  wcross_wmma_kernel<true><<<dim3(3, B), 256, 0, stream>>>(P, ws, out);
}